// Sampler_73254962201321
// MI455X (gfx1250) — compile-verified
//
#include <hip/hip_runtime.h>
#include <math.h>

// Problem constants (from reference)
#define B_ROWS   64
#define D_DIM    4096
#define V_DIM    50257
#define N_TILES  ((V_DIM + 15) / 16)   // 3142 vocab tiles
#define V_PAD    (N_TILES * 16)        // 50272 padded row stride
#define SAMPLE_KEY 42ULL

typedef __attribute__((ext_vector_type(2))) float v2f;
typedef __attribute__((ext_vector_type(4))) float v4f;
typedef __attribute__((ext_vector_type(8))) float v8f;

// ---------------------------------------------------------------------------
// Kernel 1: logits = (h @ W^T) / temperature, padded cols set to -inf.
// One wave -> one 16(M)x16(N) tile. Block = 4 waves = all 64 batch rows for
// one vocab tile, so B-matrix (W) loads are identical across waves -> WGP$ hit.
// fp32 WMMA (16x16x4), K consumed 8 at a time via b128 loads per lane:
//   lane half 0 owns k0..k0+3, lane half 1 owns k0+4..k0+7 (same bijection
//   applied to A and B, so the accumulated dot product is exact).
// ---------------------------------------------------------------------------
__global__ __launch_bounds__(128)
void gemm_logits_wmma(const float* __restrict__ h,
                      const float* __restrict__ W,
                      const float* __restrict__ temperature,
                      float* __restrict__ logits)
{
    const int lane = threadIdx.x & 31;
    const int wave = threadIdx.x >> 5;          // 0..3
    const int m0   = wave * 16;                 // batch-row tile
    const int n0   = blockIdx.x * 16;           // vocab-col tile
    const int lrow = lane & 15;
    const int koff = (lane >> 4) * 4;           // 0 or 4: which half-chunk

    const int arow = m0 + lrow;                         // h row (always valid)
    const int brow = min(n0 + lrow, V_DIM - 1);         // W row (clamped)

    const float* __restrict__ pa = h + (size_t)arow * D_DIM + koff;
    const float* __restrict__ pb = W + (size_t)brow * D_DIM + koff;

    v8f c = {};
#pragma unroll 2
    for (int k0 = 0; k0 < D_DIM; k0 += 8) {
        v4f a4 = *(const v4f*)(pa + k0);        // global_load_b128
        v4f b4 = *(const v4f*)(pb + k0);        // global_load_b128
        v2f a0; a0.x = a4.x; a0.y = a4.y;
        v2f b0; b0.x = b4.x; b0.y = b4.y;
        c = __builtin_amdgcn_wmma_f32_16x16x4_f32(false, a0, false, b0,
                                                  (short)0, c, false, false);
        v2f a1; a1.x = a4.z; a1.y = a4.w;
        v2f b1; b1.x = b4.z; b1.y = b4.w;
        c = __builtin_amdgcn_wmma_f32_16x16x4_f32(false, a1, false, b1,
                                                  (short)0, c, false, false);
    }

    // Epilogue: C/D layout: VGPR r, lanes 0-15 -> M=r, lanes 16-31 -> M=8+r;
    // N = lane & 15. Scale by rcp(temperature) (temp in [0.5,1.5) -> v_rcp_f32
    // is safe, ~1ulp) and store (-inf for padded cols).
    const int mbase = m0 + ((lane >> 4) << 3);
    const int n     = n0 + lrow;
    const bool valid = (n < V_DIM);
#pragma unroll
    for (int r = 0; r < 8; ++r) {
        const int m = mbase + r;
        const float invt = __builtin_amdgcn_rcpf(temperature[m]);
        const float val  = c[r] * invt;
        logits[(size_t)m * V_PAD + n] = valid ? val : -INFINITY;
    }
}

// ---------------------------------------------------------------------------
// Kernel 2: per-row top-p masking via threshold bisection + Gumbel-max draw.
// One 256-thread block per batch row; row (~200KB) is L2-resident (192MB L2).
// ---------------------------------------------------------------------------
__device__ __forceinline__ float block_sum(float v, float* sred, int tid)
{
    __syncthreads();
    sred[tid] = v;
    __syncthreads();
    for (int s = 128; s > 0; s >>= 1) {
        if (tid < s) sred[tid] += sred[tid + s];
        __syncthreads();
    }
    return sred[0];
}

__device__ __forceinline__ float block_max(float v, float* sred, int tid)
{
    __syncthreads();
    sred[tid] = v;
    __syncthreads();
    for (int s = 128; s > 0; s >>= 1) {
        if (tid < s) sred[tid] = fmaxf(sred[tid], sred[tid + s]);
        __syncthreads();
    }
    return sred[0];
}

__device__ __forceinline__ unsigned long long splitmix64(unsigned long long x)
{
    x += 0x9E3779B97F4A7C15ULL;
    x = (x ^ (x >> 30)) * 0xBF58476D1CE4E5B9ULL;
    x = (x ^ (x >> 27)) * 0x94D049BB133111EBULL;
    return x ^ (x >> 31);
}

__global__ __launch_bounds__(256)
void topp_sample(const float* __restrict__ logits,
                 const float* __restrict__ top_p,
                 float* __restrict__ out)
{
    __shared__ float sred[256];
    __shared__ int   sidx[256];

    const int   tid = threadIdx.x;
    const int   row = blockIdx.x;
    const float* __restrict__ L = logits + (size_t)row * V_PAD;

    // 1) row max
    float m = -INFINITY;
    for (int i = tid; i < V_PAD; i += 256) m = fmaxf(m, L[i]);
    const float rowmax = block_max(m, sred, tid);

    // 2) softmax denominator (padding exp(-inf)=0)
    float z = 0.f;
    for (int i = tid; i < V_PAD; i += 256) z += __expf(L[i] - rowmax);
    const float Z = block_sum(z, sred, tid);

    // 3) bisection for nucleus threshold tau: largest tau whose kept mass
    //    (logit >= tau) still exceeds top_p  => nucleus incl. crossing element.
    const float target = top_p[row] * Z;       // compare unnormalized masses
    float lo = rowmax - 32.0f;                 // mass(lo) ~ Z  (> target)
    float hi = rowmax + 0.5f;                  // mass(hi) = 0  (<= target)
    for (int it = 0; it < 24; ++it) {
        const float mid = 0.5f * (lo + hi);
        float ks = 0.f;
        for (int i = tid; i < V_PAD; i += 256) {
            const float l = L[i];
            if (l >= mid) ks += __expf(l - rowmax);
        }
        const float kept = block_sum(ks, sred, tid);
        if (kept > target) lo = mid; else hi = mid;
    }
    const float tau = lo;

    // 4) Gumbel-max draw over the kept set (== categorical over masked logits)
    float best    = -INFINITY;
    int   bestIdx = 0;
    for (int i = tid; i < V_PAD; i += 256) {
        const float l = L[i];
        if (l >= tau) {
            unsigned long long x =
                splitmix64((SAMPLE_KEY << 32) ^
                           ((unsigned long long)row * 0xD1342543DE82EF95ULL) ^
                           (unsigned long long)i);
            const float u = ((float)(x >> 40) + 0.5f) * (1.0f / 16777216.0f);
            const float g = l - __logf(-__logf(u));
            if (g > best) { best = g; bestIdx = i; }
        }
    }
    __syncthreads();
    sred[tid] = best;
    sidx[tid] = bestIdx;
    __syncthreads();
    for (int s = 128; s > 0; s >>= 1) {
        if (tid < s && sred[tid + s] > sred[tid]) {
            sred[tid] = sred[tid + s];
            sidx[tid] = sidx[tid + s];
        }
        __syncthreads();
    }
    if (tid == 0) out[row] = (float)sidx[0];
}

// ---------------------------------------------------------------------------
extern "C" void kernel_launch(void* const* d_in, const int* in_sizes, int n_in,
                              void* d_out, int out_size, void* d_ws, size_t ws_size,
                              hipStream_t stream)
{
    const float* h    = (const float*)d_in[0];   // [64, 4096]
    const float* W    = (const float*)d_in[1];   // [50257, 4096]
    const float* temp = (const float*)d_in[2];   // [64]
    const float* tp   = (const float*)d_in[3];   // [64]
    float*       out  = (float*)d_out;           // [64]
    float*       logits = (float*)d_ws;          // [64, V_PAD] fp32 (~12.9 MB)

    gemm_logits_wmma<<<dim3(N_TILES), dim3(128), 0, stream>>>(h, W, temp, logits);
    topp_sample<<<dim3(B_ROWS), dim3(256), 0, stream>>>(logits, tp, out);
}